// VN_Node_GNN_32796370272846
// MI455X (gfx1250) — compile-verified
//
#include <hip/hip_runtime.h>
#include <hip/hip_bf16.h>

typedef __attribute__((ext_vector_type(2))) float v2f;
typedef __attribute__((ext_vector_type(8))) float v8f;

// D = A(16x4 f32) * B(4x16 f32) + C(16x16 f32), wave32 WMMA
static __device__ inline v8f wmma4(v2f a, v2f b, v8f c) {
    return __builtin_amdgcn_wmma_f32_16x16x4_f32(
        /*neg_a=*/false, a, /*neg_b=*/false, b,
        /*c_mod=*/(short)0, c, /*reuse_a=*/false, /*reuse_b=*/false);
}

#define LDS_STRIDE 18   // 8*18 % 64 == 16 -> both lane-halves hit disjoint banks

__global__ __launch_bounds__(256) void vn_edge_kernel(
    const float* __restrict__ x,          // N x 16 x 3
    const long long* __restrict__ ei,     // 2 x E (int64)
    const float* __restrict__ ea,         // E x 1 x 3
    const float* __restrict__ nw,         // N
    const float* __restrict__ W1,         // 64 x 17
    const float* __restrict__ Wd,         // 1 x 64
    const float* __restrict__ W2,         // 64 x 64
    float* __restrict__ num,              // N x 64 x 3 (atomic accum)
    float* __restrict__ den,              // N (atomic accum)
    long long E)
{
    __shared__ float slab_all[8][64 * LDS_STRIDE];   // per-wave staging, 36.9 KB

    const int wave = threadIdx.x >> 5;
    const int lane = threadIdx.x & 31;
    const int col  = lane & 15;     // N-column / A-row within tile
    const int half = lane >> 4;     // K/row-half selector

    const long long e0 = ((long long)blockIdx.x * 8 + wave) * 16;
    if (e0 >= E) return;

    const long long e = e0 + col;
    const bool valid  = (e < E);
    const long long ec = valid ? e : 0;               // clamp; garbage masked later

    const int src = (int)ei[ec];
    const int dst = (int)ei[E + ec];
    const float w = nw[src];                          // reference: w = node_weight[src]

    const float* __restrict__ xrow = x + (long long)src * 48;
    float eav[3];
    #pragma unroll
    for (int v = 0; v < 3; ++v) eav[v] = ea[ec * 3 + v];

    // ---------------- GEMM1: m = W1(64x17) @ feat^T(17x16) per v ----------------
    v8f acc[4][3];
    #pragma unroll
    for (int mt = 0; mt < 4; ++mt)
        #pragma unroll
        for (int v = 0; v < 3; ++v) acc[mt][v] = (v8f)0.0f;

    #pragma unroll
    for (int k = 0; k < 4; ++k) {
        const int kk = k * 4 + 2 * half;              // this lane-half's K pair
        v2f B[3];
        #pragma unroll
        for (int v = 0; v < 3; ++v) {
            B[v].x = xrow[kk * 3 + v];
            B[v].y = xrow[(kk + 1) * 3 + v];
        }
        #pragma unroll
        for (int mt = 0; mt < 4; ++mt) {
            const float* w1p = W1 + (mt * 16 + col) * 17 + kk;
            v2f A; A.x = w1p[0]; A.y = w1p[1];
            #pragma unroll
            for (int v = 0; v < 3; ++v) acc[mt][v] = wmma4(A, B[v], acc[mt][v]);
        }
    }
    // K tail: channel 16 (edge_attr) via scalar FMA on D layout
    #pragma unroll
    for (int mt = 0; mt < 4; ++mt)
        #pragma unroll
        for (int r = 0; r < 8; ++r) {
            const float w1t = W1[(mt * 16 + r + 8 * half) * 17 + 16];
            #pragma unroll
            for (int v = 0; v < 3; ++v) acc[mt][v][r] += w1t * eav[v];
        }

    // ---------------- VN-ReLU (shared direction d = Wd @ m) ----------------
    float dpart[3] = {0.f, 0.f, 0.f};
    #pragma unroll
    for (int mt = 0; mt < 4; ++mt)
        #pragma unroll
        for (int r = 0; r < 8; ++r) {
            const float wd = Wd[mt * 16 + r + 8 * half];
            #pragma unroll
            for (int v = 0; v < 3; ++v) dpart[v] += wd * acc[mt][v][r];
        }
    float dv[3];
    #pragma unroll
    for (int v = 0; v < 3; ++v)
        dv[v] = dpart[v] + __shfl_xor(dpart[v], 16, 32);   // combine channel halves
    const float rinv = __frsqrt_rn(dv[0]*dv[0] + dv[1]*dv[1] + dv[2]*dv[2]);
    float du[3];
    #pragma unroll
    for (int v = 0; v < 3; ++v) du[v] = dv[v] * rinv;

    #pragma unroll
    for (int mt = 0; mt < 4; ++mt)
        #pragma unroll
        for (int r = 0; r < 8; ++r) {
            const float dp = acc[mt][0][r]*du[0] + acc[mt][1][r]*du[1] + acc[mt][2][r]*du[2];
            if (dp < 0.f) {
                #pragma unroll
                for (int v = 0; v < 3; ++v) acc[mt][v][r] -= dp * du[v];
            }
        }

    // ---------------- GEMM2: m2 = W2(64x64) @ m(64x16) per v, then scatter ------
    float* __restrict__ slab = slab_all[wave];
    #pragma unroll 1
    for (int v = 0; v < 3; ++v) {
        // D layout -> B layout via per-wave LDS (same-wave DS ops are in-order)
        #pragma unroll
        for (int mt = 0; mt < 4; ++mt)
            #pragma unroll
            for (int r = 0; r < 8; ++r)
                slab[(mt * 16 + r + 8 * half) * LDS_STRIDE + col] = acc[mt][v][r];
        __builtin_amdgcn_wave_barrier();

        v8f acc2[4];
        #pragma unroll
        for (int mt = 0; mt < 4; ++mt) acc2[mt] = (v8f)0.0f;

        #pragma unroll
        for (int k = 0; k < 16; ++k) {
            const int kk = k * 4 + 2 * half;
            v2f B; B.x = slab[kk * LDS_STRIDE + col];
                   B.y = slab[(kk + 1) * LDS_STRIDE + col];
            #pragma unroll
            for (int mt = 0; mt < 4; ++mt) {
                const float* w2p = W2 + (mt * 16 + col) * 64 + kk;
                v2f A; A.x = w2p[0]; A.y = w2p[1];
                acc2[mt] = wmma4(A, B, acc2[mt]);
            }
        }
        __builtin_amdgcn_wave_barrier();

        // weighted scatter to destination node (L2-resident f32 atomics)
        if (valid) {
            float* nb = num + (long long)dst * 192 + v;
            #pragma unroll
            for (int mt = 0; mt < 4; ++mt)
                #pragma unroll
                for (int r = 0; r < 8; ++r)
                    unsafeAtomicAdd(nb + (mt * 16 + r + 8 * half) * 3, acc2[mt][r] * w);
        }
    }
    if (valid && half == 0) unsafeAtomicAdd(den + dst, w);
}

// ---------------- node pass: h_inv -> relu MLP -> scalar out ----------------
__global__ __launch_bounds__(256) void vn_node_kernel(
    const float* __restrict__ num, const float* __restrict__ den,
    const float* __restrict__ Wo1, const float* __restrict__ bo1,
    const float* __restrict__ Wo2, const float* __restrict__ bo2,
    float* __restrict__ out, long long N)
{
    __shared__ float sW[64 * 65];    // stride-65 pad: per-lane rows hit distinct banks
    __shared__ float sb1[64];
    __shared__ float sW2[64];
    __shared__ float shinv[8][64];

    for (int i = threadIdx.x; i < 64 * 64; i += 256) sW[(i >> 6) * 65 + (i & 63)] = Wo1[i];
    if (threadIdx.x < 64) { sb1[threadIdx.x] = Wo1 ? bo1[threadIdx.x] : 0.f; sW2[threadIdx.x] = Wo2[threadIdx.x]; }
    __syncthreads();

    const int wave = threadIdx.x >> 5;
    const int lane = threadIdx.x & 31;
    const long long n = (long long)blockIdx.x * 8 + wave;
    const bool valid = (n < N);
    const long long nc = valid ? n : 0;

    const float d   = den[nc];
    const float rdn = (d > 0.f) ? (1.f / d) : 1.f;
    const float* np = num + nc * 192;

    #pragma unroll
    for (int t = 0; t < 2; ++t) {
        const int c = lane * 2 + t;
        const float h0 = np[c * 3 + 0] * rdn;
        const float h1 = np[c * 3 + 1] * rdn;
        const float h2 = np[c * 3 + 2] * rdn;
        shinv[wave][c] = sqrtf(h0 * h0 + h1 * h1 + h2 * h2 + 1e-12f);
    }
    __syncthreads();   // uniform across block

    float partial = 0.f;
    #pragma unroll
    for (int t = 0; t < 2; ++t) {
        const int j = lane * 2 + t;
        float s = sb1[j];
        #pragma unroll 8
        for (int ii = 0; ii < 64; ++ii) s += sW[j * 65 + ii] * shinv[wave][ii];
        s = fmaxf(s, 0.f);
        partial += sW2[j] * s;
    }
    #pragma unroll
    for (int off = 16; off > 0; off >>= 1) partial += __shfl_xor(partial, off, 32);
    if (valid && lane == 0) out[n] = partial + bo2[0];
}

__global__ void vn_zero_kernel(float* __restrict__ p, long long n) {
    long long i = (long long)blockIdx.x * blockDim.x + threadIdx.x;
    const long long stride = (long long)gridDim.x * blockDim.x;
    for (; i < n; i += stride) p[i] = 0.f;
}

extern "C" void kernel_launch(void* const* d_in, const int* in_sizes, int n_in,
                              void* d_out, int out_size, void* d_ws, size_t ws_size,
                              hipStream_t stream) {
    const float*     x   = (const float*)d_in[0];
    const long long* ei  = (const long long*)d_in[1];
    const float*     ea  = (const float*)d_in[2];
    const float*     nw  = (const float*)d_in[3];
    const float*     W1  = (const float*)d_in[4];
    const float*     Wd  = (const float*)d_in[5];
    const float*     W2  = (const float*)d_in[6];
    const float*     Wo1 = (const float*)d_in[7];
    const float*     bo1 = (const float*)d_in[8];
    const float*     Wo2 = (const float*)d_in[9];
    const float*     bo2 = (const float*)d_in[10];

    const long long N = in_sizes[0] / 48;   // N x 16 x 3
    const long long E = in_sizes[1] / 2;    // 2 x E

    float* num = (float*)d_ws;              // N x 64 x 3
    float* den = num + N * 192;             // N

    // zero accumulators
    vn_zero_kernel<<<2048, 256, 0, stream>>>(num, N * 192 + N);

    // edge pass: 16 edges per wave, 8 waves per block
    const long long tiles = (E + 15) / 16;
    const int edge_blocks = (int)((tiles + 7) / 8);
    vn_edge_kernel<<<edge_blocks, 256, 0, stream>>>(x, ei, ea, nw, W1, Wd, W2,
                                                    num, den, E);

    // node pass: 1 wave per node, 8 nodes per block
    const int node_blocks = (int)((N + 7) / 8);
    vn_node_kernel<<<node_blocks, 256, 0, stream>>>(num, den, Wo1, bo1, Wo2, bo2,
                                                    (float*)d_out, N);
}